// LSTM_5179730559367
// MI455X (gfx1250) — compile-verified
//
#include <hip/hip_runtime.h>
#include <math.h>

typedef __attribute__((ext_vector_type(2))) float v2f;
typedef __attribute__((ext_vector_type(8))) float v8f;

static constexpr int Bz = 32;     // batch
static constexpr int Tt = 512;    // time
static constexpr int Ff = 512;    // features
static constexpr int Hh = 1024;   // hidden
static constexpr int Cc = 1000;   // classes
static constexpr int ZK = Hh + Ff;        // 1536  (GEMM1 K)
static constexpr int GN = 4 * Hh;         // 4096  (GEMM1 N)
static constexpr int CP = 1008;           // classes padded to 63*16

// workspace layout (in floats)
static constexpr size_t WS_Z  = 0;                  // Z  [32][1536]
static constexpr size_t WS_G  = WS_Z  + (size_t)Bz * ZK;   // G  [32][4096]
static constexpr size_t WS_H  = WS_G  + (size_t)Bz * GN;   // Hl [32][1024]
static constexpr size_t WS_WT = WS_H  + (size_t)Bz * Hh;   // WT [1024][1008] = fc_w^T padded
static constexpr size_t WS_L  = WS_WT + (size_t)Hh * CP;   // L  [32][1008] logits

// ---------------- kernel 1: build Z = [h0 | data[:, T-1, :]] ----------------
__global__ void buildz_kernel(const float* __restrict__ data,
                              const float* __restrict__ h0,
                              float* __restrict__ Z) {
    int idx = blockIdx.x * blockDim.x + threadIdx.x;
    if (idx >= Bz * ZK) return;
    int b = idx / ZK, j = idx - b * ZK;
    Z[idx] = (j < Hh) ? h0[j]
                      : data[((size_t)b * Tt + (Tt - 1)) * Ff + (j - Hh)];
}

// ---------------- kernel 2: transpose fc_w (1000,1024) -> WT (1024,1008) ----
__global__ void tfc_kernel(const float* __restrict__ fcw, float* __restrict__ WT) {
    int idx = blockIdx.x * blockDim.x + threadIdx.x;
    if (idx >= Hh * CP) return;
    int k = idx / CP, n = idx - k * CP;
    WT[idx] = (n < Cc) ? fcw[(size_t)n * Hh + k] : 0.0f;
}

// ---------------- kernel 3: G[32,4096] = Z[32,1536] @ W_all (fp32 WMMA) -----
// one wave per 16-column N-tile; two 16-row M-tiles per wave; K stepped by 4.
__global__ __launch_bounds__(32)
void gates_wmma_kernel(const float* __restrict__ wf, const float* __restrict__ wi,
                       const float* __restrict__ wc, const float* __restrict__ wo,
                       const float* __restrict__ Z, float* __restrict__ G) {
    int lane = threadIdx.x;
    int hi = lane >> 4;          // half-wave select
    int nl = lane & 15;          // row (A) / column (B,C) index within tile
    int ntile = blockIdx.x;      // 0..255
    int gate  = ntile >> 6;      // which weight matrix
    int col   = (ntile & 63) << 4;
    const float* w = (gate == 0) ? wf : (gate == 1) ? wi : (gate == 2) ? wc : wo;

    v8f acc0 = {};               // rows 0..15 of G tile
    v8f acc1 = {};               // rows 16..31 of G tile

    // A: lane<16 holds K=k0,k0+1 for M=lane; lane>=16 holds K=k0+2,k0+3 for M=lane-16
    const float* zp0 = Z + (size_t)nl * ZK + 2 * hi;
    const float* zp1 = Z + (size_t)(nl + 16) * ZK + 2 * hi;
    // B: v0 = rows k0 / k0+2 striped over half-waves, v1 = rows k0+1 / k0+3
    const float* wp  = w + (size_t)(2 * hi) * Hh + col + nl;

    for (int k0 = 0; k0 < ZK; k0 += 4) {
        v2f a0, a1, bb;
        a0.x = zp0[k0];  a0.y = zp0[k0 + 1];
        a1.x = zp1[k0];  a1.y = zp1[k0 + 1];
        bb.x = wp[(size_t)k0 * Hh];
        bb.y = wp[(size_t)(k0 + 1) * Hh];
        acc0 = __builtin_amdgcn_wmma_f32_16x16x4_f32(false, a0, false, bb,
                                                     (short)0, acc0, false, false);
        acc1 = __builtin_amdgcn_wmma_f32_16x16x4_f32(false, a1, false, bb,
                                                     (short)0, acc1, false, false);
    }

    // C/D layout: VGPR v, lanes0-15 -> M=v, lanes16-31 -> M=v+8, N=lane%16
    float* gp = G + (size_t)ntile * 16 + nl;
#pragma unroll
    for (int v = 0; v < 8; ++v) {
        int m0 = v + 8 * hi;
        gp[(size_t)m0 * GN]        = acc0[v];
        gp[(size_t)(m0 + 16) * GN] = acc1[v];
    }
}

// ---------------- kernel 4: gate nonlinearities -> H_last[32,1024] ----------
__global__ void act_kernel(const float* __restrict__ G,
                           const float* __restrict__ c0,
                           float* __restrict__ Hl) {
    int idx = blockIdx.x * blockDim.x + threadIdx.x;
    if (idx >= Bz * Hh) return;
    int b = idx >> 10, k = idx & (Hh - 1);
    const float* g = G + (size_t)b * GN;
    float f  = g[k];
    float i  = g[Hh + k];
    float c1 = g[2 * Hh + k];
    float o  = g[3 * Hh + k];
    float sf = 1.0f / (1.0f + expf(-f));
    float si = 1.0f / (1.0f + expf(-i));
    float so = 1.0f / (1.0f + expf(-o));
    float c  = sf * c0[k] + si * tanhf(c1);
    Hl[idx]  = so * tanhf(c);
}

// ---------------- kernel 5: logits = relu(H_last @ WT + b) (fp32 WMMA) ------
__global__ __launch_bounds__(32)
void fc_wmma_kernel(const float* __restrict__ Hl, const float* __restrict__ WT,
                    const float* __restrict__ fcb, float* __restrict__ L) {
    int lane = threadIdx.x;
    int hi = lane >> 4, nl = lane & 15;
    int col = blockIdx.x * 16;          // 63 tiles -> 1008 padded columns

    v8f acc0 = {}, acc1 = {};
    const float* zp0 = Hl + (size_t)nl * Hh + 2 * hi;
    const float* zp1 = Hl + (size_t)(nl + 16) * Hh + 2 * hi;
    const float* wp  = WT + (size_t)(2 * hi) * CP + col + nl;

    for (int k0 = 0; k0 < Hh; k0 += 4) {
        v2f a0, a1, bb;
        a0.x = zp0[k0];  a0.y = zp0[k0 + 1];
        a1.x = zp1[k0];  a1.y = zp1[k0 + 1];
        bb.x = wp[(size_t)k0 * CP];
        bb.y = wp[(size_t)(k0 + 1) * CP];
        acc0 = __builtin_amdgcn_wmma_f32_16x16x4_f32(false, a0, false, bb,
                                                     (short)0, acc0, false, false);
        acc1 = __builtin_amdgcn_wmma_f32_16x16x4_f32(false, a1, false, bb,
                                                     (short)0, acc1, false, false);
    }

    int n = col + nl;
    float bias = (n < Cc) ? fcb[n] : 0.0f;
    float* lp = L + n;
#pragma unroll
    for (int v = 0; v < 8; ++v) {
        int m0 = v + 8 * hi;
        lp[(size_t)m0 * CP]        = fmaxf(acc0[v] + bias, 0.0f);
        lp[(size_t)(m0 + 16) * CP] = fmaxf(acc1[v] + bias, 0.0f);
    }
}

// ---------------- kernel 6: row-wise log_softmax over 1000 classes ----------
__global__ void lsm_kernel(const float* __restrict__ L, float* __restrict__ out) {
    __shared__ float red[256];
    int b = blockIdx.x, tid = threadIdx.x;
    const float* row = L + (size_t)b * CP;

    float m = -1e30f;
    for (int c = tid; c < Cc; c += 256) m = fmaxf(m, row[c]);
    red[tid] = m; __syncthreads();
    for (int s = 128; s > 0; s >>= 1) {
        if (tid < s) red[tid] = fmaxf(red[tid], red[tid + s]);
        __syncthreads();
    }
    float mx = red[0]; __syncthreads();

    float sum = 0.0f;
    for (int c = tid; c < Cc; c += 256) sum += expf(row[c] - mx);
    red[tid] = sum; __syncthreads();
    for (int s = 128; s > 0; s >>= 1) {
        if (tid < s) red[tid] += red[tid + s];
        __syncthreads();
    }
    float lse = logf(red[0]) + mx;

    for (int c = tid; c < Cc; c += 256) out[(size_t)b * Cc + c] = row[c] - lse;
}

extern "C" void kernel_launch(void* const* d_in, const int* in_sizes, int n_in,
                              void* d_out, int out_size, void* d_ws, size_t ws_size,
                              hipStream_t stream) {
    const float* data = (const float*)d_in[0];
    const float* h0   = (const float*)d_in[1];
    const float* c0   = (const float*)d_in[2];
    const float* w_f  = (const float*)d_in[3];
    const float* w_i  = (const float*)d_in[4];
    const float* w_c1 = (const float*)d_in[5];
    const float* w_o  = (const float*)d_in[6];
    const float* fc_w = (const float*)d_in[7];
    const float* fc_b = (const float*)d_in[8];
    float* out = (float*)d_out;

    float* ws = (float*)d_ws;
    float* Z  = ws + WS_Z;
    float* G  = ws + WS_G;
    float* Hl = ws + WS_H;
    float* WT = ws + WS_WT;
    float* L  = ws + WS_L;

    buildz_kernel<<<(Bz * ZK + 255) / 256, 256, 0, stream>>>(data, h0, Z);
    tfc_kernel<<<(Hh * CP + 255) / 256, 256, 0, stream>>>(fc_w, WT);
    gates_wmma_kernel<<<GN / 16, 32, 0, stream>>>(w_f, w_i, w_c1, w_o, Z, G);
    act_kernel<<<(Bz * Hh + 255) / 256, 256, 0, stream>>>(G, c0, Hl);
    fc_wmma_kernel<<<CP / 16, 32, 0, stream>>>(Hl, WT, fc_b, L);
    lsm_kernel<<<Bz, 256, 0, stream>>>(L, out);
}